// LogLikelihoodLayer_90606630077076
// MI455X (gfx1250) — compile-verified
//
#include <hip/hip_runtime.h>
#include <hip/hip_bf16.h>
#include <math.h>

typedef __attribute__((ext_vector_type(2))) float v2f;
typedef __attribute__((ext_vector_type(8))) float v8f;

// One wave (32 lanes) per (b,t) matrix. D = 64, panel width 16.
// Blocked right-looking Cholesky in LDS; trailing SYRK update via
// V_WMMA_F32_16X16X4_F32 (4 chained K=4 WMMAs per 16x16 tile, A negated).
__global__ __launch_bounds__(32)
void chol_nll_kernel(const float* __restrict__ x,
                     const float* __restrict__ mu,
                     const float* __restrict__ sigma,
                     float* __restrict__ partials) {
    const int bt   = blockIdx.x;
    const int lane = threadIdx.x;          // 0..31 (wave32)

    __shared__ float S[64 * 64];           // 16 KB matrix tile (row-major)
    __shared__ float dv[64];               // d = x - mu, overwritten by solve

    // ---- coalesced load of sigma tile into LDS (b128 per lane per iter) ----
    const float4* src = (const float4*)(sigma + (size_t)bt * 4096);
    float4* dst = (float4*)S;
    #pragma unroll
    for (int i = 0; i < 32; ++i)
        dst[i * 32 + lane] = src[i * 32 + lane];

    // d vector
    {
        const int base = bt * 64;
        dv[lane]      = x[base + lane]      - mu[base + lane];
        dv[lane + 32] = x[base + lane + 32] - mu[base + lane + 32];
    }
    __syncthreads();

    // jitter on diagonal (matches inv(sigma + 1e-8 I); logdet diff negligible)
    S[lane * 65]        += 1e-8f;
    S[(lane + 32) * 65] += 1e-8f;
    __syncthreads();

    const int n  = lane & 15;              // 0..15
    const int hi = lane >> 4;              // 0 or 1

    // ---------------- blocked Cholesky, panel width 16 ----------------
    for (int kb = 0; kb < 4; ++kb) {
        const int c0 = kb * 16;

        // (a) factor 16x16 diagonal block (unblocked, lanes 0..15 own rows)
        for (int k = 0; k < 16; ++k) {
            float pivot = sqrtf(S[(c0 + k) * 64 + c0 + k]);
            if (lane == k)                 S[(c0 + k) * 64 + c0 + k] = pivot;
            if (lane > k && lane < 16)     S[(c0 + lane) * 64 + c0 + k] /= pivot;
            __syncthreads();
            if (lane > k && lane < 16) {
                float lik = S[(c0 + lane) * 64 + c0 + k];
                for (int j = k + 1; j <= lane; ++j)
                    S[(c0 + lane) * 64 + c0 + j] -= lik * S[(c0 + j) * 64 + c0 + k];
            }
            __syncthreads();
        }

        // (b) panel solve: L21 rows r solve  L11 * l_r = a_r  (forward subst.)
        for (int r = c0 + 16 + lane; r < 64; r += 32) {
            float a[16];
            #pragma unroll
            for (int k = 0; k < 16; ++k) a[k] = S[r * 64 + c0 + k];
            #pragma unroll
            for (int k = 0; k < 16; ++k) {
                float s = a[k];
                for (int j = 0; j < k; ++j)
                    s -= a[j] * S[(c0 + k) * 64 + c0 + j];
                a[k] = s / S[(c0 + k) * 64 + c0 + k];
            }
            #pragma unroll
            for (int k = 0; k < 16; ++k) S[r * 64 + c0 + k] = a[k];
        }
        __syncthreads();

        // (c) trailing update via WMMA:  A22[ib,jb] -= Lp[ib] * Lp[jb]^T
        //     D = (-A)*B + C chained over K: 4 x (16x16x4 f32)
        for (int ib = kb + 1; ib < 4; ++ib) {
            for (int jb = kb + 1; jb <= ib; ++jb) {
                v8f c;
                #pragma unroll
                for (int v = 0; v < 8; ++v)
                    c[v] = S[(ib * 16 + v + hi * 8) * 64 + jb * 16 + n];
                #pragma unroll
                for (int kk = 0; kk < 4; ++kk) {
                    const int kbase = c0 + kk * 4 + hi * 2;
                    v2f a, b;
                    a.x = -S[(ib * 16 + n) * 64 + kbase];     // A[m][k], negated
                    a.y = -S[(ib * 16 + n) * 64 + kbase + 1];
                    b.x =  S[(jb * 16 + n) * 64 + kbase];     // B[k][n] = Lp_jb[n][k]
                    b.y =  S[(jb * 16 + n) * 64 + kbase + 1];
                    c = __builtin_amdgcn_wmma_f32_16x16x4_f32(
                            false, a, false, b, (short)0, c, false, false);
                }
                #pragma unroll
                for (int v = 0; v < 8; ++v)
                    S[(ib * 16 + v + hi * 8) * 64 + jb * 16 + n] = c[v];
            }
        }
        __syncthreads();
    }

    // ---------------- 0.5*logdet = sum(log(L_kk)) ----------------
    float ld = logf(S[lane * 65]) + logf(S[(lane + 32) * 65]);
    #pragma unroll
    for (int off = 16; off; off >>= 1)
        ld += __shfl_xor(ld, off, 32);

    // ---------------- forward substitution: y = L^{-1} d; quad = |y|^2 ----
    float quad = 0.0f;
    for (int k = 0; k < 64; ++k) {
        float yk = dv[k] / S[k * 65];      // uniform (broadcast LDS read)
        if (lane == 0) quad += yk * yk;
        for (int i = k + 1 + lane; i < 64; i += 32)
            dv[i] -= S[i * 64 + k] * yk;
        __syncthreads();
    }

    // nll contribution = 0.5*logdet + 0.5*quad = ld + 0.5*quad
    if (lane == 0) partials[bt] = ld + 0.5f * quad;
}

// Deterministic tree reduction of per-matrix partials into out[0].
__global__ __launch_bounds__(256)
void reduce_kernel(const float* __restrict__ partials, float* __restrict__ out, int n) {
    __shared__ float buf[256];
    float s = 0.0f;
    for (int i = threadIdx.x; i < n; i += 256) s += partials[i];
    buf[threadIdx.x] = s;
    __syncthreads();
    for (int off = 128; off; off >>= 1) {
        if (threadIdx.x < (unsigned)off) buf[threadIdx.x] += buf[threadIdx.x + off];
        __syncthreads();
    }
    if (threadIdx.x == 0) out[0] = buf[0];
}

extern "C" void kernel_launch(void* const* d_in, const int* in_sizes, int n_in,
                              void* d_out, int out_size, void* d_ws, size_t ws_size,
                              hipStream_t stream) {
    (void)n_in; (void)out_size; (void)ws_size;
    const float* x     = (const float*)d_in[0];
    const float* mu    = (const float*)d_in[1];
    const float* sigma = (const float*)d_in[2];
    const int BT = in_sizes[0] / 64;       // B*T = 12800

    float* partials = (float*)d_ws;        // BT floats of scratch

    chol_nll_kernel<<<BT, 32, 0, stream>>>(x, mu, sigma, partials);
    reduce_kernel<<<1, 256, 0, stream>>>(partials, (float*)d_out, BT);
}